// Digits_Caps_41764261987108
// MI455X (gfx1250) — compile-verified
//
#include <hip/hip_runtime.h>

// Problem dims
#define BN        512          // batch
#define INU       8            // in_units (k)
#define INCH      1152         // in_channels (i)
#define NJ        10           // num_units (j)
#define NU        16           // unit_size (u)
#define KKDIM     (INCH*INU)   // 9216 flattened (i,k)
#define MJU       (NJ*NU)      // 160 flattened (j,u)
#define BETA      1.45f

typedef __attribute__((ext_vector_type(2))) float v2f;
typedef __attribute__((ext_vector_type(8))) float v8f;

// ---------------------------------------------------------------------------
// xT[(i*8+k)][b] = x[b][k][i]   (gather read, coalesced write; L2 resident)
// ---------------------------------------------------------------------------
__global__ void k_transpose_x(const float* __restrict__ x, float* __restrict__ xT) {
    int idx = blockIdx.x * blockDim.x + threadIdx.x;
    if (idx >= KKDIM * BN) return;
    int b  = idx & (BN - 1);
    int kk = idx >> 9;            // / 512
    int i  = kk >> 3;
    int k  = kk & 7;
    xT[idx] = x[b * (INU * INCH) + k * INCH + i];
}

__global__ void k_zero(float* __restrict__ p, int n) {
    int i = blockIdx.x * blockDim.x + threadIdx.x;
    if (i < n) p[i] = 0.0f;
}

// ---------------------------------------------------------------------------
// c[:,j] = softmax over i (1152) of b_ij[:,j]
// ---------------------------------------------------------------------------
__global__ void k_softmax(const float* __restrict__ bij, float* __restrict__ c) {
    int j = blockIdx.x;
    int t = threadIdx.x;
    __shared__ float red[256];
    float mx = -1e30f;
    for (int i = t; i < INCH; i += 256) mx = fmaxf(mx, bij[i * NJ + j]);
    red[t] = mx; __syncthreads();
    for (int s = 128; s > 0; s >>= 1) { if (t < s) red[t] = fmaxf(red[t], red[t + s]); __syncthreads(); }
    mx = red[0]; __syncthreads();
    float sum = 0.0f;
    for (int i = t; i < INCH; i += 256) sum += __expf(bij[i * NJ + j] - mx);
    red[t] = sum; __syncthreads();
    for (int s = 128; s > 0; s >>= 1) { if (t < s) red[t] += red[t + s]; __syncthreads(); }
    float inv = 1.0f / red[0];
    for (int i = t; i < INCH; i += 256) c[i * NJ + j] = __expf(bij[i * NJ + j] - mx) * inv;
}

// ---------------------------------------------------------------------------
// Pair-packed scaled weights (A operand for the s-GEMM):
//   cWP[j][kk>>1][u][kk&1] = c[i][j] * W[i][j][u][k],  kk = i*8+k
// Output-linear thread mapping -> fully coalesced writes; one b64 per A pair.
// ---------------------------------------------------------------------------
__global__ void k_scale_w(const float* __restrict__ W, const float* __restrict__ c,
                          float* __restrict__ cWP) {
    int idx = blockIdx.x * blockDim.x + threadIdx.x;   // == output offset
    if (idx >= NJ * KKDIM * NU) return;
    int par  = idx & 1;
    int u    = (idx >> 1) & 15;
    int pair = (idx >> 5) % (KKDIM / 2);
    int j    = idx / (KKDIM * NU);
    int kk   = pair * 2 + par;
    int i    = kk >> 3;
    int k    = kk & 7;
    cWP[idx] = c[i * NJ + j] * W[i * (NJ * NU * INU) + j * (NU * INU) + u * INU + k];
}

// ---------------------------------------------------------------------------
// s-GEMM:  s[(j,u)][b] = sum_kk cWP(j,kk,u) * xT[kk][b]
// M=160 (10 j-tiles), N=512 (16 blocks x 2 b-tiles/wave), K=9216.
// 8 waves/block split over K (LDS reduce). Each wave: 2 wmma per A-b64 load,
// K unrolled x4 (8 wmma per outer iteration). EXEC all-ones (no divergence).
// ---------------------------------------------------------------------------
__global__ __launch_bounds__(256)
void k_s_gemm(const float* __restrict__ cWP, const float* __restrict__ xT,
              float* __restrict__ s) {
    int bt   = blockIdx.x * 32;                 // two 16-wide b tiles
    int j    = blockIdx.y;                      // M tile == one j
    int lane = threadIdx.x & 31;
    int wv   = threadIdx.x >> 5;                // 0..7 K-split wave id
    int m    = lane & 15;
    int h    = lane >> 4;                       // lane half -> K offset 2h

    const float* __restrict__ A = cWP + (size_t)j * (KKDIM * NU);

    v8f acc0 = {}, acc1 = {};
    for (int kk0 = wv * 16; kk0 < KKDIM; kk0 += 128) {
        // pull the next K-chunk of B toward the WGP (global_prefetch_b8)
        __builtin_prefetch(&xT[(kk0 + 2 * h + 128) * BN + bt + m], 0, 0);
        #pragma unroll
        for (int q = 0; q < 4; ++q) {
            int ra = kk0 + q * 4 + 2 * h;       // even
            v2f a  = *(const v2f*)&A[(ra >> 1) * 32 + m * 2];   // b64
            v2f b0, b1;
            b0.x = xT[ra * BN + bt + m];
            b0.y = xT[(ra + 1) * BN + bt + m];
            b1.x = xT[ra * BN + bt + 16 + m];
            b1.y = xT[(ra + 1) * BN + bt + 16 + m];
            acc0 = __builtin_amdgcn_wmma_f32_16x16x4_f32(
                       false, a, false, b0, (short)0, acc0, false, false);
            acc1 = __builtin_amdgcn_wmma_f32_16x16x4_f32(
                       false, a, false, b1, (short)0, acc1, false, false);
        }
    }

    __shared__ float red[8 * 2 * 256];          // 16 KB
    #pragma unroll
    for (int r = 0; r < 8; ++r) {
        red[(wv * 2 + 0) * 256 + (r + 8 * h) * 16 + m] = acc0[r];
        red[(wv * 2 + 1) * 256 + (r + 8 * h) * 16 + m] = acc1[r];
    }
    __syncthreads();

    int t = threadIdx.x;                        // t -> (row=t>>4, col=t&15)
    float s0 = 0.0f, s1 = 0.0f;
    #pragma unroll
    for (int w = 0; w < 8; ++w) {
        s0 += red[(w * 2 + 0) * 256 + t];
        s1 += red[(w * 2 + 1) * 256 + t];
    }
    int row = j * NU + (t >> 4);
    s[row * BN + bt + (t & 15)]      = s0;
    s[row * BN + bt + 16 + (t & 15)] = s1;
}

// ---------------------------------------------------------------------------
// squash (faithful: mag_sq sums over the num_units axis j, per (b,u))
// out_layout 0: v[(j,u)][b] for the T-GEMM; 1: out[b][j][u] (final output)
// ---------------------------------------------------------------------------
__global__ void k_squash(const float* __restrict__ s, float* __restrict__ v,
                         int out_layout) {
    int idx = blockIdx.x * blockDim.x + threadIdx.x;   // over BN*NU
    if (idx >= BN * NU) return;
    int b = idx & (BN - 1);
    int u = idx >> 9;
    float sv[NJ];
    float msq = 0.0f;
    #pragma unroll
    for (int j = 0; j < NJ; ++j) {
        float t = s[(j * NU + u) * BN + b];
        sv[j] = t;
        msq += t * t;
    }
    float fac = msq / ((BETA + msq) * sqrtf(msq));
    if (out_layout == 0) {
        #pragma unroll
        for (int j = 0; j < NJ; ++j) v[(j * NU + u) * BN + b] = fac * sv[j];
    } else {
        #pragma unroll
        for (int j = 0; j < NJ; ++j) v[(b * NJ + j) * NU + u] = fac * sv[j];
    }
}

// ---------------------------------------------------------------------------
// T-GEMM:  T[kk][(j,u)] = sum_b xT[kk][b] * v[(j,u)][b]
// M=9216 (576 tiles, 4 waves/block), N=160 (10 tiles), K=512 unrolled x4.
// Both operands are contiguous b64 loads per lane.
// ---------------------------------------------------------------------------
__global__ __launch_bounds__(128)
void k_t_gemm(const float* __restrict__ xT, const float* __restrict__ v,
              float* __restrict__ T) {
    int lane = threadIdx.x & 31;
    int wv   = threadIdx.x >> 5;                // 0..3
    int kkt  = (blockIdx.x * 4 + wv) * 16;
    int jut  = blockIdx.y * 16;
    int m    = lane & 15;
    int h    = lane >> 4;

    v8f acc = {};
    for (int b0 = 0; b0 < BN; b0 += 16) {
        #pragma unroll
        for (int q = 0; q < 4; ++q) {
            int bq = b0 + q * 4 + 2 * h;
            v2f a  = *(const v2f*)&xT[(kkt + m) * BN + bq];  // A[m][bq..bq+1]
            v2f bb = *(const v2f*)&v [(jut + m) * BN + bq];  // B[bq..][jut+m]
            acc = __builtin_amdgcn_wmma_f32_16x16x4_f32(
                      false, a, false, bb, (short)0, acc, false, false);
        }
    }
    #pragma unroll
    for (int r = 0; r < 8; ++r)
        T[(size_t)(kkt + r + 8 * h) * MJU + jut + m] = acc[r];
}

// ---------------------------------------------------------------------------
// b_ij[i][j] = (1/B) * sum_{u,k} W[i][j][u][k] * T[(i*8+k)][(j*16+u)]
// ---------------------------------------------------------------------------
__global__ void k_b_update(const float* __restrict__ W, const float* __restrict__ T,
                           float* __restrict__ bij) {
    int idx = blockIdx.x * blockDim.x + threadIdx.x;   // over INCH*NJ
    if (idx >= INCH * NJ) return;
    int j = idx % NJ;
    int i = idx / NJ;
    float acc = 0.0f;
    for (int u = 0; u < NU; ++u) {
        #pragma unroll
        for (int k = 0; k < INU; ++k)
            acc += W[i * (NJ * NU * INU) + j * (NU * INU) + u * INU + k]
                 * T[(size_t)(i * INU + k) * MJU + j * NU + u];
    }
    bij[idx] = acc * (1.0f / BN);
}

// ---------------------------------------------------------------------------
extern "C" void kernel_launch(void* const* d_in, const int* in_sizes, int n_in,
                              void* d_out, int out_size, void* d_ws, size_t ws_size,
                              hipStream_t stream) {
    const float* x = (const float*)d_in[0];   // (512, 8, 1152)
    const float* W = (const float*)d_in[1];   // (1152, 10, 16, 8)
    float* out = (float*)d_out;               // (512, 10, 4, 4) == (b, j, u)

    float* ws   = (float*)d_ws;
    float* xT   = ws;                              // 9216*512   = 4,718,592
    float* cWP  = xT  + (size_t)KKDIM * BN;        // 10*9216*16 = 1,474,560
    float* sbuf = cWP + (size_t)NJ * KKDIM * NU;   // 160*512    =    81,920
    float* vbuf = sbuf + (size_t)MJU * BN;         // 160*512    =    81,920
    float* Tbuf = vbuf + (size_t)MJU * BN;         // 9216*160   = 1,474,560
    float* bij  = Tbuf + (size_t)KKDIM * MJU;      // 1152*10    =    11,520
    float* cbuf = bij  + (size_t)INCH * NJ;        // 1152*10    =    11,520
    // total ~7.85M floats ~31.4 MB

    k_transpose_x<<<(KKDIM * BN + 255) / 256, 256, 0, stream>>>(x, xT);
    k_zero<<<(INCH * NJ + 255) / 256, 256, 0, stream>>>(bij, INCH * NJ);

    for (int it = 0; it < 3; ++it) {
        k_softmax<<<NJ, 256, 0, stream>>>(bij, cbuf);
        k_scale_w<<<(NJ * KKDIM * NU + 255) / 256, 256, 0, stream>>>(W, cbuf, cWP);
        k_s_gemm<<<dim3(BN / 32, NJ), 256, 0, stream>>>(cWP, xT, sbuf);
        if (it < 2) {
            k_squash<<<(BN * NU + 255) / 256, 256, 0, stream>>>(sbuf, vbuf, 0);
            k_t_gemm<<<dim3(KKDIM / 64, NJ), 128, 0, stream>>>(xT, vbuf, Tbuf);
            k_b_update<<<(INCH * NJ + 255) / 256, 256, 0, stream>>>(W, Tbuf, bij);
        } else {
            k_squash<<<(BN * NU + 255) / 256, 256, 0, stream>>>(sbuf, out, 1);
        }
    }
}